// CurriculumMasking_43576738185859
// MI455X (gfx1250) — compile-verified
//
#include <hip/hip_runtime.h>
#include <math.h>

// CurriculumMasking for B=4,H=16,S=1024 (fp32).
// One workgroup per row; weights live in registers across all phases;
// noise row staged via gfx1250 async global->LDS (ASYNCcnt) overlapped
// with the entropy reduction. Memory-bound: ~12KB/row, one touch per byte.
// Entropy accumulated in log2 domain so v_log_f32 is used raw (no branchy
// libm expansion); streaming loads/stores are nontemporal (128-bit).

#define S_LEN 1024
#define TPB   256          // 8 waves of 32
#define NWAVE (TPB / 32)

#define EPS_F          1e-8f
#define BASE_MASK_PROB 0.15f
#define ENTROPY_TARGET 0.7f
#define LN2_F          0.6931471805599453f
#define LOG_S          6.9314718055994531f   // ln(1024)
#define NEG_LOG_EPS    18.420680743952367f   // -ln(1e-8)

typedef __attribute__((ext_vector_type(4))) float v4f;   // native vector: OK for nontemporal builtins

__device__ __forceinline__ float wave_sum(float v) {
#pragma unroll
    for (int o = 16; o > 0; o >>= 1) v += __shfl_xor(v, o, 32);
    return v;
}

__device__ __forceinline__ void wave_argmax(float& v, int& i) {
#pragma unroll
    for (int o = 16; o > 0; o >>= 1) {
        float ov = __shfl_xor(v, o, 32);
        int   oi = __shfl_xor(i, o, 32);
        if (ov > v || (ov == v && oi < i)) { v = ov; i = oi; }
    }
}

// w * log2(w), xlogy-safe, branch-free: raw v_log_f32 + v_cndmask.
__device__ __forceinline__ float xlog2x(float w) {
    float t = w * __builtin_amdgcn_logf(w);   // v_log_f32 (base-2)
    return (w > 0.f) ? t : 0.f;
}

__global__ void __launch_bounds__(TPB)
curriculum_mask_kernel(const float* __restrict__ weights,
                       const float* __restrict__ noise,
                       float* __restrict__ out_w,
                       float* __restrict__ out_entropy,
                       float* __restrict__ out_maskrate,
                       float* __restrict__ out_target) {
    const int    row    = blockIdx.x;
    const int    tid    = threadIdx.x;
    const size_t rowoff = (size_t)row * S_LEN;

    __shared__ float s_noise[S_LEN];          // 4 KB staged noise row
    __shared__ float s_a[NWAVE], s_b[NWAVE], s_c[NWAVE];
    __shared__ int   s_i[NWAVE];

    // ---- Phase 0: kick off async global->LDS copy of the noise row ----
    // Each lane copies 16B from global memory at nbase plus the offset in voff to the LDS address in lds_dst (GVS mode).
    // Tracked by ASYNCcnt; overlaps with the entropy reduction below.
    {
        unsigned lds_dst = (unsigned)(size_t)(&s_noise[0]) + (unsigned)(tid * 16);
        unsigned voff    = (unsigned)(tid * 16);
        const float* nbase = noise + rowoff;
        asm volatile("global_load_async_to_lds_b128 %0, %1, %2"
                     :
                     : "v"(lds_dst), "v"(voff), "s"(nbase)
                     : "memory");
    }

    // ---- Phase 1: load weights (stay in registers), entropy reduction ----
    // Read-once stream: nontemporal so it doesn't displace L2.
    v4f w4 = __builtin_nontemporal_load((const v4f*)(weights + rowoff) + tid);

    float lsum = (w4.x + w4.y) + (w4.z + w4.w);
    float lwlw = ((xlog2x(w4.x) + xlog2x(w4.y)) +
                  (xlog2x(w4.z) + xlog2x(w4.w)));       // sum of w*log2(w)

    float mv = w4.x; int mi = tid * 4;
    if (w4.y > mv) { mv = w4.y; mi = tid * 4 + 1; }
    if (w4.z > mv) { mv = w4.z; mi = tid * 4 + 2; }
    if (w4.w > mv) { mv = w4.w; mi = tid * 4 + 3; }

    float sum = wave_sum(lsum);
    float wlw = wave_sum(lwlw);
    wave_argmax(mv, mi);

    const int wave = tid >> 5, lane = tid & 31;
    if (lane == 0) { s_a[wave] = sum; s_b[wave] = wlw; s_c[wave] = mv; s_i[wave] = mi; }
    __syncthreads();

    float rs = 0.f, rw = 0.f, rmv = -1.f; int rmi = 0;
#pragma unroll
    for (int i = 0; i < NWAVE; ++i) {
        rs += s_a[i]; rw += s_b[i];
        float v = s_c[i]; int ix = s_i[i];
        if (v > rmv || (v == rmv && ix < rmi)) { rmv = v; rmi = ix; }
    }
    __syncthreads();   // s_a/s_b reused in phase 2

    // ---- Row statistics (log2 domain -> nats via one ln2 multiply) ----
    const bool  degen  = (rs < EPS_F);
    const float inv_rs = degen ? 0.f : __builtin_amdgcn_rcpf(rs);  // v_rcp_f32
    if (degen) rmi = 0;                                  // uniform row -> top-1 = idx 0
    float entropy = degen
        ? LOG_S
        : fminf(LN2_F * (__builtin_amdgcn_logf(rs) - rw * inv_rs), NEG_LOG_EPS);
    float norm_e    = fminf(fmaxf(entropy * (1.f / LOG_S), 0.f), 1.f);
    float keep_prob = 1.f - BASE_MASK_PROB * (1.f - norm_e);

    // normalized weights (registers)
    const float uni = 1.f / (float)S_LEN;
    float wn0 = degen ? uni : w4.x * inv_rs;
    float wn1 = degen ? uni : w4.y * inv_rs;
    float wn2 = degen ? uni : w4.z * inv_rs;
    float wn3 = degen ? uni : w4.w * inv_rs;

    // ---- Phase 2: consume async-staged noise, build mask ----
    asm volatile("s_wait_asynccnt 0" ::: "memory");      // this wave's async loads done
    __syncthreads();                                     // all waves' data visible

    v4f n4 = ((const v4f*)s_noise)[tid];
    float m0 = (n4.x < keep_prob) ? 1.f : 0.f;
    float m1 = (n4.y < keep_prob) ? 1.f : 0.f;
    float m2 = (n4.z < keep_prob) ? 1.f : 0.f;
    float m3 = (n4.w < keep_prob) ? 1.f : 0.f;

    float lact = (m0 + m1) + (m2 + m3);
    float lms  = wn0 * m0 + wn1 * m1 + wn2 * m2 + wn3 * m3;

    float act = wave_sum(lact);
    float msw = wave_sum(lms);
    if (lane == 0) { s_a[wave] = act; s_b[wave] = msw; }
    __syncthreads();

    float ract = 0.f, rms = 0.f;
#pragma unroll
    for (int i = 0; i < NWAVE; ++i) { ract += s_a[i]; rms += s_b[i]; }

    // ---- min_active fallback (k = 1): empty rows keep their argmax ----
    const bool needs_more = (ract < 1.f);
    float msum, mask_rate;
    if (needs_more) {
        const int c0 = tid * 4;
        m0 = (c0     == rmi) ? 1.f : 0.f;
        m1 = (c0 + 1 == rmi) ? 1.f : 0.f;
        m2 = (c0 + 2 == rmi) ? 1.f : 0.f;
        m3 = (c0 + 3 == rmi) ? 1.f : 0.f;
        msum      = degen ? uni : rmv * inv_rs;
        mask_rate = 1.f - uni;
    } else {
        msum      = rms;
        mask_rate = 1.f - ract * uni;
    }

    // ---- renormalize (fall back to wn if masked sum ~ 0) ----
    const bool  valid = (msum > EPS_F);
    const float scale = valid ? __builtin_amdgcn_rcpf(msum) : 1.f;
    v4f o;
    o.x = valid ? wn0 * m0 * scale : wn0;
    o.y = valid ? wn1 * m1 * scale : wn1;
    o.z = valid ? wn2 * m2 * scale : wn2;
    o.w = valid ? wn3 * m3 * scale : wn3;
    // Write-once stream, never re-read: nontemporal store.
    __builtin_nontemporal_store(o, (v4f*)(out_w + rowoff) + tid);

    if (tid == 0) {
        out_entropy[row]  = entropy;
        out_maskrate[row] = mask_rate;
        out_target[row]   = LOG_S * ENTROPY_TARGET;
    }
}

extern "C" void kernel_launch(void* const* d_in, const int* in_sizes, int n_in,
                              void* d_out, int out_size, void* d_ws, size_t ws_size,
                              hipStream_t stream) {
    (void)n_in; (void)out_size; (void)d_ws; (void)ws_size;

    const float* weights = (const float*)d_in[0];
    const float* noise   = (const float*)d_in[1];

    const size_t fw_elems = (size_t)in_sizes[0];        // B*H*S*S
    const int    rows     = (int)(fw_elems / S_LEN);    // B*H*S

    float* out_w        = (float*)d_out;
    float* out_entropy  = out_w + fw_elems;
    float* out_maskrate = out_entropy + rows;
    float* out_target   = out_maskrate + rows;

    curriculum_mask_kernel<<<dim3(rows), dim3(TPB), 0, stream>>>(
        weights, noise, out_w, out_entropy, out_maskrate, out_target);
}